// MultiheadAttention_20925080666277
// MI455X (gfx1250) — compile-verified
//
#include <hip/hip_runtime.h>
#include <hip/hip_bf16.h>
#include <math.h>

// Problem constants (match reference)
#define LQ 512   // sequence length
#define NB 4     // batch
#define EE 512   // embed dim
#define HH 8     // heads
#define DD 64    // head dim

#define AW 640   // A-cat width in attention kernel (E + 2*D)

typedef float v2f __attribute__((ext_vector_type(2)));
typedef float v8f __attribute__((ext_vector_type(8)));

// D = A(16x4) * B(4x16) + C, f32 WMMA (wave32).
// A frag: lane l -> row m=l&15, a[v] = A[m][2*(l>>4)+v]
// B frag: lane l -> col n=l&15, b[v] = B[2*(l>>4)+v][n]
// C/D:    lane l, reg r -> C[r + 8*(l>>4)][l&15]
static __device__ __forceinline__ v8f wmma4(v2f a, v2f b, v8f c) {
  return __builtin_amdgcn_wmma_f32_16x16x4_f32(false, a, false, b, (short)0, c,
                                               false, false);
}

// ---------------------------------------------------------------------------
// Kernel A: fused QKV projection  Y = X @ W^T + bias, scatter to q/kT/v layouts
//   X  : (2048, 512)  query rows r = i*B + b
//   W  : (1536, 512)
//   Qh : (B, L, E)   q * 0.125
//   KT : (B, E, L)   k transposed per batch
//   VA : (B, L, E)   v
// ---------------------------------------------------------------------------
__global__ __launch_bounds__(128) void qkv_proj_kernel(
    const float* __restrict__ X, const float* __restrict__ W,
    const float* __restrict__ bias, float* __restrict__ Qh,
    float* __restrict__ KT, float* __restrict__ VA) {
  const int lane = threadIdx.x & 31;
  const int wave = threadIdx.x >> 5;
  const int tile = blockIdx.x * 4 + wave;     // 128 x 96 tiles
  const int rowT = tile / 96;
  const int colT = tile - rowT * 96;
  const int r0 = rowT * 16, c0 = colT * 16;
  const int half = lane >> 4;
  const int mn = lane & 15;

  const float* Arow = X + (size_t)(r0 + mn) * EE;  // A row m
  const float* Brow = W + (size_t)(c0 + mn) * EE;  // B col n == W row n

  v8f acc = {0.f, 0.f, 0.f, 0.f, 0.f, 0.f, 0.f, 0.f};
  for (int kk = 0; kk < EE; kk += 4) {
    const int k0 = kk + half * 2;
    v2f a, b;
    a[0] = Arow[k0]; a[1] = Arow[k0 + 1];
    b[0] = Brow[k0]; b[1] = Brow[k0 + 1];
    acc = wmma4(a, b, acc);
  }
  for (int r = 0; r < 8; ++r) {
    const int m = r + half * 8;
    const int R = r0 + m;         // 0..2047
    const int C = c0 + mn;        // 0..1535
    const int i = R >> 2;         // R = i*B + b
    const int b = R & 3;
    const float v = acc[r] + bias[C];
    if (C < EE) {
      Qh[((size_t)b * LQ + i) * EE + C] = v * 0.125f;  // 1/sqrt(64)
    } else if (C < 2 * EE) {
      const int e = C - EE;
      KT[((size_t)b * EE + e) * LQ + i] = v;
    } else {
      const int e = C - 2 * EE;
      VA[((size_t)b * LQ + i) * EE + e] = v;
    }
  }
}

// ---------------------------------------------------------------------------
// Kernel B: attention core. One workgroup per (b, query-pair i0=2p, i1=2p+1);
// 256 threads = 8 waves. All 16 WMMA rows used (8 heads x 2 queries).
//
//  A-cat(16,640): rows 0-7  = [Qbd_i0(8x512) | q_i0(8x64) |    0     ]
//                 rows 8-15 = [Qbd_i1(8x512) |    0       | q_i1(8x64)]
//  B-cat(640,512) = [KT(512x512) ; RK_i0(64x512) ; RK_i1(64x512)]
//  S(16,512) = A-cat @ B-cat   -> row m = head (m&7) of query i0+(m>>3)
//  softmax rows
//  O(16,640) = S @ [V(512x512) | RV_i0(512x64) | RV_i1(512x64)]
//  out[i0+sel][h][d] = O[h+8*sel][h*64+d] + O[h+8*sel][512 + sel*64 + d]
// ---------------------------------------------------------------------------
__global__ __launch_bounds__(256) void attn_core_kernel(
    const float* __restrict__ Qh, const float* __restrict__ KT,
    const float* __restrict__ VA, const float* __restrict__ RK,
    const float* __restrict__ RV, float* __restrict__ AO) {
  extern __shared__ float lds[];
  float* ldsA = lds;             // 16*640 floats: A-cat, then reused as O
  float* ldsS = lds + 16 * AW;   // 16*512 floats: scores -> weights

  const int tid = threadIdx.x;
  const int lane = tid & 31;
  const int wave = tid >> 5;  // 0..7
  const int half = lane >> 4;
  const int mn = lane & 15;

  const int b = blockIdx.x >> 8;          // 0..3
  const int ip = blockIdx.x & 255;        // query pair
  const int i0 = ip * 2;

  const float* q0 = Qh + ((size_t)b * LQ + i0) * EE;
  const float* q1 = q0 + EE;
  const float* ktb = KT + (size_t)b * EE * LQ;
  const float* vab = VA + (size_t)b * LQ * EE;
  const float* rk0 = RK + ((size_t)b * LQ + i0) * (LQ * DD);  // (64,512) view
  const float* rk1 = rk0 + (size_t)LQ * DD;
  const float* rv0 = RV + ((size_t)b * LQ + i0) * (LQ * DD);  // (512,64)
  const float* rv1 = rv0 + (size_t)LQ * DD;

  // Build A-cat (16 x 640)
  for (int idx = tid; idx < 16 * AW; idx += 256) {
    const int m = idx / AW;
    const int c = idx - m * AW;
    const int h = m & 7;
    const int sel = m >> 3;
    const float* qr = sel ? q1 : q0;
    float v;
    if (c < EE)            v = ((c >> 6) == h) ? qr[c] : 0.0f;
    else if (c < EE + DD)  v = (sel == 0) ? qr[h * DD + (c - EE)] : 0.0f;
    else                   v = (sel == 1) ? qr[h * DD + (c - EE - DD)] : 0.0f;
    ldsA[idx] = v;
  }
  __syncthreads();

  // Phase 1: scores S(16,512)
  for (int t = wave; t < 32; t += 8) {
    const int n = t * 16 + mn;
    v8f acc = {0.f, 0.f, 0.f, 0.f, 0.f, 0.f, 0.f, 0.f};
    for (int kk = 0; kk < EE; kk += 4) {  // content rows: K^T
      const int k0 = kk + half * 2;
      v2f a, bf;
      a[0] = ldsA[mn * AW + k0]; a[1] = ldsA[mn * AW + k0 + 1];
      bf[0] = ktb[(size_t)k0 * LQ + n];
      bf[1] = ktb[(size_t)(k0 + 1) * LQ + n];
      acc = wmma4(a, bf, acc);
    }
    for (int kk = 0; kk < DD; kk += 4) {  // relation_k rows of i0
      const int k0 = kk + half * 2;
      v2f a, bf;
      a[0] = ldsA[mn * AW + EE + k0]; a[1] = ldsA[mn * AW + EE + k0 + 1];
      bf[0] = rk0[(size_t)k0 * LQ + n];
      bf[1] = rk0[(size_t)(k0 + 1) * LQ + n];
      acc = wmma4(a, bf, acc);
    }
    for (int kk = 0; kk < DD; kk += 4) {  // relation_k rows of i1
      const int k0 = kk + half * 2;
      v2f a, bf;
      a[0] = ldsA[mn * AW + EE + DD + k0]; a[1] = ldsA[mn * AW + EE + DD + k0 + 1];
      bf[0] = rk1[(size_t)k0 * LQ + n];
      bf[1] = rk1[(size_t)(k0 + 1) * LQ + n];
      acc = wmma4(a, bf, acc);
    }
    for (int r = 0; r < 8; ++r) {
      ldsS[(r + 8 * half) * 512 + n] = acc[r];
    }
  }
  __syncthreads();

  // Phase 2: softmax; 16 rows over 8 waves (wave32 shuffles)
  for (int row = wave; row < 16; row += 8) {
    float mx = -3.4e38f;
    for (int j = lane; j < LQ; j += 32) mx = fmaxf(mx, ldsS[row * 512 + j]);
    for (int off = 16; off > 0; off >>= 1) mx = fmaxf(mx, __shfl_xor(mx, off, 32));
    float sum = 0.f;
    for (int j = lane; j < LQ; j += 32) {
      const float e = __expf(ldsS[row * 512 + j] - mx);
      ldsS[row * 512 + j] = e;
      sum += e;
    }
    for (int off = 16; off > 0; off >>= 1) sum += __shfl_xor(sum, off, 32);
    const float inv = (sum > 0.f) ? 1.0f / sum : 0.0f;  // NaN guard
    for (int j = lane; j < LQ; j += 32) ldsS[row * 512 + j] *= inv;
  }
  __syncthreads();

  // Phase 3: O(16,640) = S(16,512) @ [V | RV_i0 | RV_i1]; 40 N-tiles
  for (int t = wave; t < 40; t += 8) {
    v8f acc = {0.f, 0.f, 0.f, 0.f, 0.f, 0.f, 0.f, 0.f};
    for (int kk = 0; kk < LQ; kk += 4) {
      const int k0 = kk + half * 2;
      v2f a, bf;
      a[0] = ldsS[mn * 512 + k0]; a[1] = ldsS[mn * 512 + k0 + 1];
      if (t < 32) {  // wave-uniform branches
        const int n = t * 16 + mn;
        bf[0] = vab[(size_t)k0 * EE + n];
        bf[1] = vab[(size_t)(k0 + 1) * EE + n];
      } else if (t < 36) {
        const int d = (t - 32) * 16 + mn;
        bf[0] = rv0[(size_t)k0 * DD + d];
        bf[1] = rv0[(size_t)(k0 + 1) * DD + d];
      } else {
        const int d = (t - 36) * 16 + mn;
        bf[0] = rv1[(size_t)k0 * DD + d];
        bf[1] = rv1[(size_t)(k0 + 1) * DD + d];
      }
      acc = wmma4(a, bf, acc);
    }
    const int col = t * 16 + mn;
    for (int r = 0; r < 8; ++r) {
      ldsA[(r + 8 * half) * AW + col] = acc[r];  // O overwrites A-cat
    }
  }
  __syncthreads();

  // Epilogue: two output rows, AO layout (L, B, E)
  for (int idx = tid; idx < 2 * EE; idx += 256) {
    const int sel = idx >> 9;       // which query of the pair
    const int e = idx & 511;
    const int h = e >> 6, d = e & 63;
    const int row = h + 8 * sel;
    const float v = ldsA[row * AW + e] + ldsA[row * AW + EE + sel * DD + d];
    AO[((size_t)(i0 + sel) * NB + b) * EE + e] = v;
  }
}

// ---------------------------------------------------------------------------
// Kernel C: output projection  Y = AO @ Wo^T + bias   (2048x512)@(512x512)
// ---------------------------------------------------------------------------
__global__ __launch_bounds__(128) void out_proj_kernel(
    const float* __restrict__ AO, const float* __restrict__ Wo,
    const float* __restrict__ bias, float* __restrict__ Y) {
  const int lane = threadIdx.x & 31;
  const int wave = threadIdx.x >> 5;
  const int tile = blockIdx.x * 4 + wave;  // 128 x 32 tiles
  const int rowT = tile >> 5;
  const int colT = tile & 31;
  const int r0 = rowT * 16, c0 = colT * 16;
  const int half = lane >> 4;
  const int mn = lane & 15;

  const float* Arow = AO + (size_t)(r0 + mn) * EE;
  const float* Brow = Wo + (size_t)(c0 + mn) * EE;

  v8f acc = {0.f, 0.f, 0.f, 0.f, 0.f, 0.f, 0.f, 0.f};
  for (int kk = 0; kk < EE; kk += 4) {
    const int k0 = kk + half * 2;
    v2f a, b;
    a[0] = Arow[k0]; a[1] = Arow[k0 + 1];
    b[0] = Brow[k0]; b[1] = Brow[k0 + 1];
    acc = wmma4(a, b, acc);
  }
  for (int r = 0; r < 8; ++r) {
    const int R = r0 + r + half * 8;
    const int C = c0 + mn;
    Y[(size_t)R * EE + C] = acc[r] + bias[C];
  }
}

extern "C" void kernel_launch(void* const* d_in, const int* in_sizes, int n_in,
                              void* d_out, int out_size, void* d_ws,
                              size_t ws_size, hipStream_t stream) {
  (void)in_sizes; (void)n_in; (void)out_size; (void)ws_size;
  const float* query = (const float*)d_in[0];       // (L,B,E)
  const float* rel_k = (const float*)d_in[1];       // (B,L,L,D)
  const float* rel_v = (const float*)d_in[2];       // (B,L,L,D)
  const float* w_in  = (const float*)d_in[3];       // (3E,E)
  const float* b_in  = (const float*)d_in[4];       // (3E)
  const float* w_out = (const float*)d_in[5];       // (E,E)
  const float* b_out = (const float*)d_in[6];       // (E)
  float* out = (float*)d_out;                       // (L,B,E)

  float* ws = (float*)d_ws;
  const size_t SEG = (size_t)NB * LQ * EE;  // 1M floats
  float* Qh = ws;             // (B,L,E)
  float* KT = ws + SEG;       // (B,E,L)
  float* VA = ws + 2 * SEG;   // (B,L,E)
  float* AO = ws + 3 * SEG;   // (L,B,E)

  // 128x96 tiles, 4 waves/block
  qkv_proj_kernel<<<3072, 128, 0, stream>>>(query, w_in, b_in, Qh, KT, VA);
  // one workgroup per (b, query-pair); dynamic LDS = 16*640 + 16*512 floats
  const size_t ldsBytes = (size_t)(16 * AW + 16 * 512) * sizeof(float);
  attn_core_kernel<<<NB * (LQ / 2), 256, ldsBytes, stream>>>(Qh, KT, VA, rel_k,
                                                             rel_v, AO);
  // 128x32 tiles, 4 waves/block
  out_proj_kernel<<<1024, 128, 0, stream>>>(AO, w_out, b_out, out);
}